// LowLatencyTwoShotAllReduce_5875515261130
// MI455X (gfx1250) — compile-verified
//
#include <hip/hip_runtime.h>
#include <stdint.h>

#define TP 8
#define TOKENS 2048
#define HIDDEN 7168
#define EPS_F 0.0078125f

#define THREADS 256
#define VEC4_PER_ROW (HIDDEN / 4)        // 1792 vec4 per row
#define CHUNKS (VEC4_PER_ROW / THREADS)  // 7 vec4 per thread

typedef __attribute__((ext_vector_type(4))) float v4f;  // native vector (builtin-legal)
typedef __attribute__((ext_vector_type(4))) int v4i;
typedef __attribute__((address_space(1))) v4i as1_v4i;  // global int4
typedef __attribute__((address_space(3))) v4i as3_v4i;  // LDS int4

// Async global->LDS b128 copy (CDNA5 ASYNCcnt path). Per-lane 16B copy.
// Signature (probe-confirmed): (global int4*, lds int4*, imm offset, imm cpol)
__device__ __forceinline__ void async_copy_b128(const void* gsrc, void* ldst) {
    __builtin_amdgcn_global_load_async_to_lds_b128(
        (as1_v4i*)gsrc, (as3_v4i*)ldst, /*offset=*/0, /*cpol=*/0);
}

__device__ __forceinline__ void wait_async_zero() {
#if __has_builtin(__builtin_amdgcn_s_wait_asynccnt)
    __builtin_amdgcn_s_wait_asynccnt(0);
#else
    asm volatile("s_wait_asynccnt 0" ::: "memory");
#endif
}

// Streaming (read-once / write-once) helpers: non-temporal TH policy so the
// ~646MB/call stream does not thrash the 192MB L2 (keeps gamma resident).
__device__ __forceinline__ v4f load_nt(const v4f* p) {
    return __builtin_nontemporal_load(p);
}
__device__ __forceinline__ void store_nt(v4f* p, v4f v) {
    __builtin_nontemporal_store(v, p);
}

__global__ __launch_bounds__(THREADS) void twoshot_allreduce_rmsnorm_kernel(
    const float* __restrict__ x,            // [TP, TOKENS, HIDDEN]
    const float* __restrict__ residual_in,  // [TOKENS, HIDDEN]
    const float* __restrict__ gamma,        // [HIDDEN]
    float* __restrict__ out)                // [normed | residual_out]
{
    __shared__ __align__(16) float s_res[HIDDEN];
    __shared__ __align__(16) float s_gamma[HIDDEN];
    __shared__ float s_red[THREADS / 32];
    __shared__ float s_inv;

    const int t = threadIdx.x;
    const int token = blockIdx.x;

    const v4f* res4 = (const v4f*)residual_in + (size_t)token * VEC4_PER_ROW;
    const v4f* gam4 = (const v4f*)gamma;
    v4f* s_res4 = (v4f*)s_res;
    v4f* s_gam4 = (v4f*)s_gamma;

    // ---- Phase 0: async prefetch residual row + gamma into LDS (ASYNCcnt) ----
    // gamma keeps default (RT) policy: 28KB reused by all 2048 blocks -> L2-resident.
#pragma unroll
    for (int j = 0; j < CHUNKS; ++j) {
        const int idx = j * THREADS + t;
        async_copy_b128(res4 + idx, s_res4 + idx);
        async_copy_b128(gam4 + idx, s_gam4 + idx);
    }

    // ---- Phase 1: 8-way rank sum in registers (global_load_b128, NT) ----
    const size_t rank_stride4 = (size_t)TOKENS * VEC4_PER_ROW;
    const v4f* xbase = (const v4f*)x + (size_t)token * VEC4_PER_ROW;
    v4f acc[CHUNKS];
#pragma unroll
    for (int j = 0; j < CHUNKS; ++j) {
        const int idx = j * THREADS + t;
        v4f a = load_nt(xbase + idx);
#pragma unroll
        for (int r = 1; r < TP; ++r) {
            a += load_nt(xbase + (size_t)r * rank_stride4 + idx);
        }
        acc[j] = a;
    }

    // Drain our wave's async copies, then sync so all waves' LDS data is visible.
    wait_async_zero();
    __syncthreads();

    // ---- Phase 2: residual add, write residual_out (NT), accumulate sum-of-squares ----
    float ss = 0.0f;
    v4f* resout4 = (v4f*)(out + (size_t)TOKENS * HIDDEN) + (size_t)token * VEC4_PER_ROW;
#pragma unroll
    for (int j = 0; j < CHUNKS; ++j) {
        const int idx = j * THREADS + t;
        v4f a = acc[j] + s_res4[idx];
        acc[j] = a;
        ss += a.x * a.x + a.y * a.y + a.z * a.z + a.w * a.w;
        store_nt(resout4 + idx, a);
    }

    // ---- Phase 3: block reduction of sum-of-squares (wave32 shuffles + LDS) ----
#pragma unroll
    for (int m = 16; m >= 1; m >>= 1)
        ss += __shfl_xor(ss, m, 32);
    const int wave = t >> 5;
    const int lane = t & 31;
    if (lane == 0) s_red[wave] = ss;
    __syncthreads();
    if (t == 0) {
        float tot = 0.0f;
#pragma unroll
        for (int w = 0; w < THREADS / 32; ++w) tot += s_red[w];
        s_inv = rsqrtf(tot / (float)HIDDEN + EPS_F);
    }
    __syncthreads();
    const float inv = s_inv;

    // ---- Phase 4: scale by gamma (LDS) and write normed output (NT) ----
    v4f* norm4 = (v4f*)out + (size_t)token * VEC4_PER_ROW;
#pragma unroll
    for (int j = 0; j < CHUNKS; ++j) {
        const int idx = j * THREADS + t;
        const v4f o = acc[j] * inv * s_gam4[idx];
        store_nt(norm4 + idx, o);
    }
}

extern "C" void kernel_launch(void* const* d_in, const int* in_sizes, int n_in,
                              void* d_out, int out_size, void* d_ws, size_t ws_size,
                              hipStream_t stream) {
    (void)in_sizes; (void)n_in; (void)out_size; (void)d_ws; (void)ws_size;
    const float* x           = (const float*)d_in[0];
    const float* residual_in = (const float*)d_in[1];
    const float* gamma       = (const float*)d_in[2];
    float* out               = (float*)d_out;
    twoshot_allreduce_rmsnorm_kernel<<<TOKENS, THREADS, 0, stream>>>(
        x, residual_in, gamma, out);
}